// FusionGAT3_27204322853480
// MI455X (gfx1250) — compile-verified
//
#include <hip/hip_runtime.h>
#include <hip/hip_bf16.h>

#define NN     3000
#define NP     3008           // padded to multiple of 16
#define TILES  (NP / 16)      // 188
#define NFEAT  512
#define NHID1  64
#define NHID2  32
#define FUSD   64
#define NCLS   8
#define NHEADS 4
#define NADJ   2
#define AH     (NADJ * NHEADS)
#define ALPHA  0.2f

typedef __attribute__((ext_vector_type(2))) float v2f;
typedef __attribute__((ext_vector_type(8))) float v8f;

__device__ __forceinline__ float eluf(float v) {
    return v > 0.f ? v : (__expf(v) - 1.f);
}

// ---------------------------------------------------------------------------
// Wh[ah][n][o] = X[n][:] @ W[ah][:][o]   (f32 WMMA 16x16x4, K-step 4)
// Templated dims -> pointer-increment loads with immediate offsets.
// ---------------------------------------------------------------------------
template <int F, int O>
__global__ void gemm_wh(const float* __restrict__ X, int nrowsX,
                        const float* __restrict__ W,   // [AH][F][O]
                        float* __restrict__ Wh) {      // [AH][NP][O]
    int wid  = threadIdx.x >> 5;
    int lane = threadIdx.x & 31;
    int t    = blockIdx.x;                 // row tile
    int ah   = blockIdx.y;
    int row  = lane & 15;                  // also output column-in-tile
    int half = lane >> 4;
    int o0   = wid * 16;

    int nr = t * 16 + row;
    if (nr > nrowsX - 1) nr = nrowsX - 1;  // clamp padded rows

    const float* xp = X + (size_t)nr * F + 2 * half;
    const float* wp = W + (size_t)ah * F * O + (size_t)(2 * half) * O + o0 + row;

    v8f acc = (v8f)(0.0f);
    for (int k = 0; k < F; k += 4) {
        v2f a = *(const v2f*)xp;           // A: row nr, K = k+2h, k+2h+1
        v2f b;
        b.x = wp[0];                       // B: K rows, col o0+row
        b.y = wp[O];
        acc = __builtin_amdgcn_wmma_f32_16x16x4_f32(
                  false, a, false, b, (short)0, acc, false, false);
        xp += 4;
        wp += 4 * O;
    }

    float* out = Wh + (size_t)ah * NP * O;
#pragma unroll
    for (int r = 0; r < 8; r++) {
        int m = t * 16 + r + 8 * half;     // always < NP
        out[(size_t)m * O + o0 + row] = acc[r];
    }
}

// ---------------------------------------------------------------------------
// f1/f2 = Wh . a[:O] / a[O:]
// ---------------------------------------------------------------------------
__global__ void fvec_kernel(const float* __restrict__ Wh,
                            const float* __restrict__ avec,  // [AH][2O]
                            float* __restrict__ f1, float* __restrict__ f2, int O) {
    int idx = blockIdx.x * blockDim.x + threadIdx.x;
    if (idx >= AH * NP) return;
    int ah = idx / NP, n = idx % NP;
    const float* w  = Wh + ((size_t)ah * NP + n) * O;
    const float* av = avec + (size_t)ah * 2 * O;
    float s1 = 0.f, s2 = 0.f;
    for (int o = 0; o < O; o++) { float v = w[o]; s1 += v * av[o]; s2 += v * av[O + o]; }
    f1[idx] = s1; f2[idx] = s2;
}

// ---------------------------------------------------------------------------
// One m-tile of the flash attention inner loop.
//   adjp/f2p : per-lane pointers already offset by +2*half
//   lb       : per-lane LDS pointer = buf + (2*half)*OS + row
// GUARD=true only for the final tile (m >= NN masking).
// ---------------------------------------------------------------------------
template <int O, bool GUARD>
__device__ __forceinline__ void flash_tile(int mt, int twoHalf,
                                           const int* __restrict__ adjp,
                                           const float* __restrict__ f2p,
                                           const float* __restrict__ lb,
                                           float f1n, v8f (&acc)[O / 16], float& rs) {
    constexpr int OT = O / 16;
    constexpr int OS = O + 8;              // padded LDS row stride (bank-split halves)
    int m0 = mt * 16;
    const int*   ap = adjp + m0;
    const float* fp = f2p + m0;

    v2f P[4];
#pragma unroll
    for (int kk = 0; kk < 4; kk++) {
        int2 av;
        if (GUARD) {
            int m = m0 + 4 * kk + twoHalf;
            av.x = (m     < NN) ? ap[4 * kk]     : 0;
            av.y = (m + 1 < NN) ? ap[4 * kk + 1] : 0;
        } else {
            av = *(const int2*)(ap + 4 * kk);
        }
        v2f f2x = *(const v2f*)(fp + 4 * kk);   // f2 padded to NP: always safe
#pragma unroll
        for (int j = 0; j < 2; j++) {
            float e = f1n + (j ? f2x.y : f2x.x);
            e = e > 0.f ? e : ALPHA * e;        // leaky relu
            e = fminf(e, 70.f);                 // overflow guard (max-free softmax)
            float ex = __expf(e);
            float p  = ((j ? av.y : av.x) > 0) ? ex : 0.f;
            if (j == 0) P[kk].x = p; else P[kk].y = p;
            rs += p;
        }
    }
#pragma unroll
    for (int ct = 0; ct < OT; ct++) {
#pragma unroll
        for (int kk = 0; kk < 4; kk++) {
            v2f b;
            b.x = lb[4 * kk * OS + ct * 16];        // K = 4kk+2h   (LDS)
            b.y = lb[(4 * kk + 1) * OS + ct * 16];  // K = 4kk+2h+1 (LDS)
            acc[ct] = __builtin_amdgcn_wmma_f32_16x16x4_f32(
                          false, P[kk], false, b, (short)0, acc[ct], false, false);
        }
    }
}

// ---------------------------------------------------------------------------
// Flash-style masked GAT attention: hp = elu(softmax(mask(lrelu(f1+f2))) @ Wh)
// One wave per 16-row tile; Wh m-tiles double-buffered in LDS via async copy
// (shared by the block's 4 waves -> 4x less global B traffic).
// ---------------------------------------------------------------------------
template <int O>
__global__ __launch_bounds__(128) void flash_gat(
        const int*   __restrict__ adj,   // [NADJ][NN][NN]
        const float* __restrict__ Wh,    // [AH][NP][O]
        const float* __restrict__ f1v,   // [AH][NP]
        const float* __restrict__ f2v,   // [AH][NP]
        float*       __restrict__ hp) {  // [NADJ][NP][NHEADS*O]
    constexpr int OT = O / 16;
    constexpr int OS = O + 8;
    __shared__ float lbuf[2][16 * OS];

    int tid  = threadIdx.x;
    int wid  = tid >> 5;
    int lane = tid & 31;
    int t    = blockIdx.x * 4 + wid;     // row tile 0..187
    int ah   = blockIdx.y;
    int a    = ah / NHEADS, h = ah % NHEADS;
    int row  = lane & 15;
    int half = lane >> 4;
    int twoHalf = 2 * half;

    int n  = t * 16 + row;
    int nc = n < NN ? n : NN - 1;
    const int*   adjp = adj + ((size_t)a * NN + nc) * NN + twoHalf;
    const float  f1n  = f1v[ah * NP + nc];
    const float* f2p  = f2v + (size_t)ah * NP + twoHalf;
    const float* whp  = Wh + (size_t)ah * NP * O;
    const float* lb0  = &lbuf[0][twoHalf * OS + row];
    const float* lb1  = &lbuf[1][twoHalf * OS + row];

    // async-stage one 16xO Wh tile (contiguous 16*O floats) into padded LDS rows
    auto stage = [&](int b, int mt) {
        const char* g = (const char*)(whp + (size_t)mt * 16 * O);
        unsigned lbase = (unsigned)(uintptr_t)&lbuf[b][0];
        constexpr int CPR = O / 4;                 // 16B chunks per row
#pragma unroll
        for (int c = tid; c < 16 * CPR; c += 128) {
            int r = c / CPR, cc = c % CPR;
            unsigned loff = (unsigned)((r * OS + cc * 4) * 4);
            unsigned long long ga = (unsigned long long)(uintptr_t)(g + (r * O + cc * 4) * 4);
            asm volatile("global_load_async_to_lds_b128 %0, %1, off"
                         :: "v"(lbase + loff), "v"(ga) : "memory");
        }
    };
    auto waitsync = [&]() {
        asm volatile("s_wait_asynccnt 0x0" ::: "memory");
        __syncthreads();
    };

    v8f acc[OT];
#pragma unroll
    for (int ct = 0; ct < OT; ct++) acc[ct] = (v8f)(0.0f);
    float rs = 0.f;

    stage(0, 0);
    waitsync();
    for (int mt = 0; mt < TILES - 2; mt += 2) {
        stage(1, mt + 1);
        flash_tile<O, false>(mt, twoHalf, adjp, f2p, lb0, f1n, acc, rs);
        waitsync();
        stage(0, mt + 2);
        flash_tile<O, false>(mt + 1, twoHalf, adjp, f2p, lb1, f1n, acc, rs);
        waitsync();
    }
    // tiles 186 (buf0, staged by last loop iteration) and 187 (buf1, guarded)
    stage(1, TILES - 1);
    flash_tile<O, false>(TILES - 2, twoHalf, adjp, f2p, lb0, f1n, acc, rs);
    waitsync();
    flash_tile<O, true>(TILES - 1, twoHalf, adjp, f2p, lb1, f1n, acc, rs);

    // combine half-row partial sums: lanes L and L^16 cover all 16 columns of row L%16
    rs += __shfl_xor(rs, 16);

#pragma unroll
    for (int r = 0; r < 8; r++) {
        float denom = __shfl(rs, r + 8 * half);        // rowsum for output row r+8h
        int node = t * 16 + r + 8 * half;
        if (node < NN) {
#pragma unroll
            for (int ct = 0; ct < OT; ct++) {
                float v = acc[ct][r] / denom;
                hp[((size_t)a * NP + node) * (NHEADS * O) + h * O + ct * 16 + row] = eluf(v);
            }
        }
    }
}

// ---------------------------------------------------------------------------
// g[a][n][j] = elu( hp[a][n][:] . wint[j][:] + bint[j] )
// ---------------------------------------------------------------------------
__global__ void int_kernel(const float* __restrict__ hp, const float* __restrict__ wint,
                           const float* __restrict__ bint, float* __restrict__ g,
                           int Fin, int Fout) {
    int idx = blockIdx.x * blockDim.x + threadIdx.x;
    int total = NADJ * NN * Fout;
    if (idx >= total) return;
    int j = idx % Fout;
    int n = (idx / Fout) % NN;
    int a = idx / (Fout * NN);
    const float* hr = hp + ((size_t)a * NP + n) * Fin;
    const float* wr = wint + (size_t)j * Fin;
    float s = bint[j];
    for (int f = 0; f < Fin; f++) s += hr[f] * wr[f];
    g[((size_t)a * NP + n) * Fout + j] = eluf(s);
}

// ---------------------------------------------------------------------------
// out[n][j] = bfus[j] + sum_a sum_c g[a][n][c] * wfus[j][a*C+c]
// ---------------------------------------------------------------------------
__global__ void fus_kernel(const float* __restrict__ g, const float* __restrict__ wfus,
                           const float* __restrict__ bfus, float* __restrict__ out,
                           int C, int Fout) {
    int idx = blockIdx.x * blockDim.x + threadIdx.x;
    if (idx >= NN * Fout) return;
    int j = idx % Fout, n = idx / Fout;
    float s = bfus[j];
    for (int a = 0; a < NADJ; a++) {
        const float* gr = g + ((size_t)a * NP + n) * C;
        const float* wr = wfus + (size_t)j * (NADJ * C) + a * C;
        for (int c = 0; c < C; c++) s += gr[c] * wr[c];
    }
    out[(size_t)n * Fout + j] = s;
}

// ---------------------------------------------------------------------------
// row-wise log_softmax over NCLS, into d_out
// ---------------------------------------------------------------------------
__global__ void logsoftmax_kernel(const float* __restrict__ logits, float* __restrict__ out) {
    int n = blockIdx.x * blockDim.x + threadIdx.x;
    if (n >= NN) return;
    const float* z = logits + (size_t)n * NCLS;
    float m = z[0];
    for (int c = 1; c < NCLS; c++) m = fmaxf(m, z[c]);
    float s = 0.f;
    for (int c = 0; c < NCLS; c++) s += __expf(z[c] - m);
    float ls = logf(s);
    for (int c = 0; c < NCLS; c++) out[(size_t)n * NCLS + c] = z[c] - m - ls;
}

// ---------------------------------------------------------------------------
// scalar: mean|wfus1| + mean|wfus2|  -> d_out[NN*NCLS]
// ---------------------------------------------------------------------------
__global__ void l1_kernel(const float* __restrict__ wfus1, const float* __restrict__ wfus2,
                          float* __restrict__ out) {
    __shared__ float sh[256];
    const int n1 = FUSD * FUSD * NADJ;      // 8192
    const int n2 = NCLS * NCLS * NADJ;      // 128
    float s = 0.f;
    for (int i = threadIdx.x; i < n1; i += 256) s += fabsf(wfus1[i]) * (1.0f / n1);
    for (int i = threadIdx.x; i < n2; i += 256) s += fabsf(wfus2[i]) * (1.0f / n2);
    sh[threadIdx.x] = s;
    __syncthreads();
    for (int st = 128; st > 0; st >>= 1) {
        if (threadIdx.x < st) sh[threadIdx.x] += sh[threadIdx.x + st];
        __syncthreads();
    }
    if (threadIdx.x == 0) out[NN * NCLS] = sh[0];
}

// ---------------------------------------------------------------------------
extern "C" void kernel_launch(void* const* d_in, const int* in_sizes, int n_in,
                              void* d_out, int out_size, void* d_ws, size_t ws_size,
                              hipStream_t stream) {
    const float* x     = (const float*)d_in[0];
    const int*   adj   = (const int*)  d_in[1];
    const float* W1    = (const float*)d_in[2];
    const float* a1    = (const float*)d_in[3];
    const float* W2    = (const float*)d_in[4];
    const float* a2    = (const float*)d_in[5];
    const float* wint1 = (const float*)d_in[6];
    const float* bint1 = (const float*)d_in[7];
    const float* wfus1 = (const float*)d_in[8];
    const float* bfus1 = (const float*)d_in[9];
    const float* wint2 = (const float*)d_in[10];
    const float* bint2 = (const float*)d_in[11];
    const float* wfus2 = (const float*)d_in[12];
    const float* bfus2 = (const float*)d_in[13];
    float* out = (float*)d_out;

    // workspace partition (floats)
    float* p = (float*)d_ws;
    float* Wh1  = p; p += (size_t)AH * NP * NHID1;
    float* f11  = p; p += (size_t)AH * NP;
    float* f21  = p; p += (size_t)AH * NP;
    float* hp1  = p; p += (size_t)NADJ * NP * (NHEADS * NHID1);
    float* g1   = p; p += (size_t)NADJ * NP * FUSD;
    float* out1 = p; p += (size_t)NP * FUSD;
    float* Wh2  = p; p += (size_t)AH * NP * NHID2;
    float* f12  = p; p += (size_t)AH * NP;
    float* f22  = p; p += (size_t)AH * NP;
    float* hp2  = p; p += (size_t)NADJ * NP * (NHEADS * NHID2);
    float* g2   = p; p += (size_t)NADJ * NP * NCLS;
    float* lgt  = p; p += (size_t)NP * NCLS;

    // ---- stage 1 ----
    gemm_wh<NFEAT, NHID1><<<dim3(TILES, AH), 32 * (NHID1 / 16), 0, stream>>>(x, NN, W1, Wh1);
    fvec_kernel<<<(AH * NP + 255) / 256, 256, 0, stream>>>(Wh1, a1, f11, f21, NHID1);
    flash_gat<NHID1><<<dim3(TILES / 4, AH), 128, 0, stream>>>(adj, Wh1, f11, f21, hp1);
    int_kernel<<<(NADJ * NN * FUSD + 255) / 256, 256, 0, stream>>>(
        hp1, wint1, bint1, g1, NHEADS * NHID1, FUSD);
    fus_kernel<<<(NN * FUSD + 255) / 256, 256, 0, stream>>>(g1, wfus1, bfus1, out1, FUSD, FUSD);

    // ---- stage 2 ----
    gemm_wh<FUSD, NHID2><<<dim3(TILES, AH), 32 * (NHID2 / 16), 0, stream>>>(out1, NN, W2, Wh2);
    fvec_kernel<<<(AH * NP + 255) / 256, 256, 0, stream>>>(Wh2, a2, f12, f22, NHID2);
    flash_gat<NHID2><<<dim3(TILES / 4, AH), 128, 0, stream>>>(adj, Wh2, f12, f22, hp2);
    int_kernel<<<(NADJ * NN * NCLS + 255) / 256, 256, 0, stream>>>(
        hp2, wint2, bint2, g2, NHEADS * NHID2, NCLS);
    fus_kernel<<<(NN * NCLS + 255) / 256, 256, 0, stream>>>(g2, wfus2, bfus2, lgt, NCLS, NCLS);

    // ---- outputs ----
    logsoftmax_kernel<<<(NN + 255) / 256, 256, 0, stream>>>(lgt, out);
    l1_kernel<<<1, 256, 0, stream>>>(wfus1, wfus2, out);
}